// ReTrend_PolynomialRegression_73830487818327
// MI455X (gfx1250) — compile-verified
//
#include <hip/hip_runtime.h>
#include <hip/hip_bf16.h>

typedef __attribute__((ext_vector_type(2))) float v2f;
typedef __attribute__((ext_vector_type(8))) float v8f;

#define TDIM 8192
#define NNODES 4096
#define TM 8   // 16-row tiles per wave along M (reuse B/C across these)

__global__ __launch_bounds__(256)
void retrend_poly_wmma(const float* __restrict__ w, float* __restrict__ out) {
    const int lane = threadIdx.x & 31;
    const int wave = threadIdx.x >> 5;

    const int nTilesN   = NNODES / 16;                 // 256
    const int waveId    = blockIdx.x * 8 + wave;       // 0 .. 16383
    const int tileN     = waveId & (nTilesN - 1);
    const int tileMgrp  = waveId / nTilesN;            // 0 .. 63
    const int n0        = tileN * 16;
    const int m0        = tileMgrp * (16 * TM);

    const int colInTile = lane & 15;
    const int col       = n0 + colInTile;
    const int hi        = lane >> 4;                   // 0: K=0,1  1: K=2,3

    // ---- B matrix (4x16 f32): lane half selects K-pair, column = lane&15 ----
    v2f b;
    b.x = w[(size_t)(hi * 2 + 0) * NNODES + col];      // weights[0 or 2] rows (K=0 or K=2)
    b.y = w[(size_t)(hi * 2 + 1) * NNODES + col];

    // ---- C matrix: broadcast bias w0[col] into all 8 accumulator VGPRs ----
    const float w0 = w[col];
    v8f c;
#pragma unroll
    for (int i = 0; i < 8; ++i) c[i] = w0;

#pragma unroll
    for (int t = 0; t < TM; ++t) {
        const int mt = m0 + t * 16;

        // ---- A matrix (16x4 f32): row j = mt + (lane&15); lanes 0-15 hold
        // (j, j^2) for K=0,1; lanes 16-31 hold (j^3, j^4) for K=2,3 ----
        const float j  = (float)(mt + colInTile);
        const float j2 = j * j;
        const float j3 = j2 * j;
        const float j4 = j2 * j2;
        v2f a;
        a.x = hi ? j3 : j;    // branch-free: EXEC stays all-ones for WMMA
        a.y = hi ? j4 : j2;

        // D = A x B + C  -> v_wmma_f32_16x16x4_f32
        v8f d = __builtin_amdgcn_wmma_f32_16x16x4_f32(
            /*neg_a=*/false, a, /*neg_b=*/false, b,
            /*c_mod=*/(short)0, c, /*reuse_a=*/false, /*reuse_b=*/false);

        // ---- Store: VGPR v holds D[mt + hi*8 + v][col] ----
        float* base = out + (size_t)(mt + hi * 8) * NNODES + col;
#pragma unroll
        for (int v = 0; v < 8; ++v)
            base[(size_t)v * NNODES] = d[v];
    }
}

extern "C" void kernel_launch(void* const* d_in, const int* in_sizes, int n_in,
                              void* d_out, int out_size, void* d_ws, size_t ws_size,
                              hipStream_t stream) {
    // d_in[0] = unshifted_nodes (T*NUM_NODES f32) -- unused by the reference math
    // d_in[1] = weights ((DEG+1)*NUM_NODES f32)
    const float* w   = (const float*)d_in[1];
    float*       out = (float*)d_out;

    // waves = (T/16/TM) * (NUM_NODES/16) = 64 * 256 = 16384; 8 waves per block
    const int totalWaves = (TDIM / 16 / TM) * (NNODES / 16);
    const int blocks     = totalWaves / 8;   // 2048
    retrend_poly_wmma<<<blocks, 256, 0, stream>>>(w, out);
}